// Grid4D_84688165142533
// MI455X (gfx1250) — compile-verified
//
#include <hip/hip_runtime.h>

// Grid geometry (compile-time constants from the reference):
//   xyuv_grid: [1, R=8, X=9, Y=9, U=512, V=512] float32, V fastest.
#define R_CH 8
#define GX 9
#define GY 9
#define GU 512
#define GV 512

// Byte strides within the grid.
#define STRIDE_R ((size_t)GX * GY * GU * GV * 4)   // 84,934,656 B per R-slab
#define STRIDE_X ((unsigned)(GY * GU * GV * 4))    // 9,437,184
#define STRIDE_Y ((unsigned)(GU * GV * 4))         // 1,048,576
#define STRIDE_U ((unsigned)(GV * 4))              // 2,048

// Clang ext-vector (not HIP_vector_type) so the nontemporal builtins accept it.
typedef float f4v __attribute__((ext_vector_type(4)));

__global__ __launch_bounds__(256) void grid4d_sample_kernel(
    const float* __restrict__ xyuv,
    const float* __restrict__ grid,
    const float* __restrict__ xyuv_min,
    const float* __restrict__ xyuv_max,
    float* __restrict__ out,
    int n_pts)
{
    const int n = blockIdx.x * blockDim.x + threadIdx.x;
    if (n >= n_pts) return;

    // Touch-once coord stream: non-temporal so it does not evict grid lines
    // from the 192 MB L2 (the grid has ~12x line reuse, the coords have none).
    const f4v c = __builtin_nontemporal_load(
        reinterpret_cast<const f4v*>(xyuv) + n);

    // pos = (c - min) * (size-1)/(max - min)   (align_corners=True).
    // The divisors are wave-uniform; v_rcp_f32 (exact for the actual
    // power-of-two range 1.0) + FMA instead of four IEEE division chains
    // keeps the prologue off the gather critical path.
    const float sx = (float)(GX - 1) * __builtin_amdgcn_rcpf(xyuv_max[0] - xyuv_min[0]);
    const float sy = (float)(GY - 1) * __builtin_amdgcn_rcpf(xyuv_max[1] - xyuv_min[1]);
    const float su = (float)(GU - 1) * __builtin_amdgcn_rcpf(xyuv_max[2] - xyuv_min[2]);
    const float sv = (float)(GV - 1) * __builtin_amdgcn_rcpf(xyuv_max[3] - xyuv_min[3]);

    const float px = (c.x - xyuv_min[0]) * sx;
    const float py = (c.y - xyuv_min[1]) * sy;
    const float pu = (c.z - xyuv_min[2]) * su;
    const float pv = (c.w - xyuv_min[3]) * sv;

    // i0 = floor(pos), clamped to [0, size-2]; frac = pos - i0.
    // For pos == size-1 this is numerically identical to the reference's
    // OOB-masked corner (full weight lands on index size-1 either way).
    int ix = (int)floorf(px); ix = ix < 0 ? 0 : (ix > GX - 2 ? GX - 2 : ix);
    int iy = (int)floorf(py); iy = iy < 0 ? 0 : (iy > GY - 2 ? GY - 2 : iy);
    int iu = (int)floorf(pu); iu = iu < 0 ? 0 : (iu > GU - 2 ? GU - 2 : iu);
    int iv = (int)floorf(pv); iv = iv < 0 ? 0 : (iv > GV - 2 ? GV - 2 : iv);

    const float fx = px - (float)ix;
    const float fy = py - (float)iy;
    const float fu = pu - (float)iu;
    const float fv = pv - (float)iv;

    // Byte offset of the base corner inside one R-slab (fits in 32 bits).
    const unsigned base = (unsigned)ix * STRIDE_X + (unsigned)iy * STRIDE_Y
                        + (unsigned)iu * STRIDE_U + (unsigned)iv * 4u;

    // 8 (x,y,u)-corner offsets; the V-pair is one 8-byte load.
    // k bits: [2]=x corner, [1]=y corner, [0]=u corner.
    unsigned off[8];
    #pragma unroll
    for (int k = 0; k < 8; ++k) {
        off[k] = base + ((k & 4) ? STRIDE_X : 0u)
                      + ((k & 2) ? STRIDE_Y : 0u)
                      + ((k & 1) ? STRIDE_U : 0u);
    }

    const char* gB = reinterpret_cast<const char*>(grid);

    // gfx1250 global_prefetch_b8: pull the first two R-slabs' corner lines
    // toward the cache while the remaining address/weight VALU retires.
    #pragma unroll
    for (int k = 0; k < 8; ++k) {
        __builtin_prefetch(gB + off[k]);
        __builtin_prefetch(gB + STRIDE_R + off[k]);
    }

    float acc[R_CH];
    // Unroll 4 -> up to 32 b64 gathers in flight per lane between waits:
    // this kernel is bound by random-gather latency, not VALU.
    #pragma unroll 4
    for (int r = 0; r < R_CH; ++r) {
        const char* g = gB + (size_t)r * STRIDE_R;
        float q[8];
        #pragma unroll
        for (int k = 0; k < 8; ++k) {
            // Two adjacent V samples; merges into one global_load_b64
            // (verified in the round-1 histogram).
            const float* gp = reinterpret_cast<const float*>(g + off[k]);
            const float p0 = gp[0];
            const float p1 = gp[1];
            q[k] = fmaf(fv, p1 - p0, p0);            // lerp along V
        }
        // lerp along U
        const float s0 = fmaf(fu, q[1] - q[0], q[0]);  // (x0,y0)
        const float s1 = fmaf(fu, q[3] - q[2], q[2]);  // (x0,y1)
        const float s2 = fmaf(fu, q[5] - q[4], q[4]);  // (x1,y0)
        const float s3 = fmaf(fu, q[7] - q[6], q[6]);  // (x1,y1)
        // lerp along Y
        const float t0 = fmaf(fy, s1 - s0, s0);        // x0
        const float t1 = fmaf(fy, s3 - s2, s2);        // x1
        // lerp along X
        acc[r] = fmaf(fx, t1 - t0, t0);
    }

    // Each lane owns a contiguous 32-byte output row -> the wave stores 1 KB
    // contiguous. Output is write-once: non-temporal to spare L2 for the grid.
    f4v* o = reinterpret_cast<f4v*>(out + (size_t)n * R_CH);
    const f4v lo = {acc[0], acc[1], acc[2], acc[3]};
    const f4v hi = {acc[4], acc[5], acc[6], acc[7]};
    __builtin_nontemporal_store(lo, o);
    __builtin_nontemporal_store(hi, o + 1);
}

extern "C" void kernel_launch(void* const* d_in, const int* in_sizes, int n_in,
                              void* d_out, int out_size, void* d_ws, size_t ws_size,
                              hipStream_t stream) {
    const float* xyuv = (const float*)d_in[0];   // [N, 4]
    const float* grid = (const float*)d_in[1];   // [1, 8, 9, 9, 512, 512]
    const float* mn   = (const float*)d_in[2];   // [4]
    const float* mx   = (const float*)d_in[3];   // [4]
    float* out = (float*)d_out;                  // [N, 8]

    const int n_pts = in_sizes[0] / 4;
    const int block = 256;                        // 8 wave32 waves per block
    const int nblocks = (n_pts + block - 1) / block;

    hipLaunchKernelGGL(grid4d_sample_kernel, dim3(nblocks), dim3(block), 0, stream,
                       xyuv, grid, mn, mx, out, n_pts);
}